// ComponentAwa_36721970381319
// MI455X (gfx1250) — compile-verified
//
#include <hip/hip_runtime.h>

typedef __attribute__((ext_vector_type(2))) float v2f;
typedef __attribute__((ext_vector_type(8))) float v8f;

#define H 128
#define W 128

// ================== Phase 1: depthwise towers + gate-group reduction ==================
// grid: (8 row-blocks, 4 gates, 32 batch), block 256.
// Each WG: loop over the gate-group's channels; for each, load a 26-row halo slab,
// run the 5 depthwise 3x3 layers fully in LDS (valid region shrinks 26->16 rows),
// accumulate the surviving 16x128 tile into an LDS gate accumulator. No atomics.
__global__ __launch_bounds__(256)
void dw_gate_kernel(const float* __restrict__ lm,
                    const float* __restrict__ w_lm,
                    const float* __restrict__ b_lm,
                    float* __restrict__ gates)
{
    const int blk = blockIdx.x;   // row block 0..7
    const int g   = blockIdx.y;   // gate 0..3 (eye, nose, jaw, mouth)
    const int b   = blockIdx.z;   // batch 0..31
    const int tid = threadIdx.x;
    const int y0  = blk * 16;

    __shared__ float buf[2][26][138];   // rows with 5-col zero pads each side
    __shared__ float gacc[16][128];

    for (int idx = tid; idx < 16*128; idx += 256)
        gacc[idx >> 7][idx & 127] = 0.f;
    // zero the x-halo pad columns (0..4, 133..137) in both buffers; never rewritten
    for (int idx = tid; idx < 2*26*10; idx += 256) {
        int bsel = idx / 260, rem = idx % 260;
        int i = rem / 10, j = rem % 10;
        int col = (j < 5) ? j : (133 + (j - 5));
        buf[bsel][i][col] = 0.f;
    }
    __syncthreads();

    const int s0[4]    = {17, 27, 0, 48};
    const int cnt0[4]  = {10,  9, 17, 20};
    const int segs = (g == 0) ? 2 : 1;   // eye = [17,27) + [36,48)

    for (int s = 0; s < segs; ++s) {
        const int cstart = (s == 0) ? s0[g] : 36;
        const int ccnt   = (s == 0) ? cnt0[g] : 12;
        for (int ci = 0; ci < ccnt; ++ci) {
            const int c = cstart + ci;
            const float* src = lm + ((size_t)(b*68 + c) * H) * W;
            // load rows y0-5 .. y0+20 (zero outside the image)
            for (int idx = tid; idx < 26*128; idx += 256) {
                int i = idx >> 7, x = idx & 127;
                int y = y0 - 5 + i;
                buf[0][i][x + 5] = (y >= 0 && y < H) ? src[y*W + x] : 0.f;
            }
            __syncthreads();
            // 5 depthwise layers, ping-pong; final result lands in buf[1]
            for (int k = 0; k < 5; ++k) {
                const float* wp = w_lm + (size_t)(k*68 + c) * 9;   // uniform -> s_load
                const float w00=wp[0],w01=wp[1],w02=wp[2],
                            w10=wp[3],w11=wp[4],w12=wp[5],
                            w20=wp[6],w21=wp[7],w22=wp[8];
                const float bias = b_lm[k*68 + c];
                float (*sb)[138] = buf[k & 1];
                float (*db)[138] = buf[(k + 1) & 1];
                const int rlo = k + 1, nrows = (24 - k) - rlo + 1;
                for (int idx = tid; idx < nrows*128; idx += 256) {
                    int i = rlo + (idx >> 7);
                    int x = idx & 127;
                    int y = y0 - 5 + i;
                    float v;
                    if (y < 0 || y >= H) {
                        v = 0.f;  // implicit zero padding between layers
                    } else {
                        int xc = x + 5;
                        v = bias
                          + w00*sb[i-1][xc-1] + w01*sb[i-1][xc] + w02*sb[i-1][xc+1]
                          + w10*sb[i  ][xc-1] + w11*sb[i  ][xc] + w12*sb[i  ][xc+1]
                          + w20*sb[i+1][xc-1] + w21*sb[i+1][xc] + w22*sb[i+1][xc+1];
                    }
                    db[i][x + 5] = v;
                }
                __syncthreads();
            }
            // accumulate valid rows (i = 5..20 -> y0..y0+15)
            for (int idx = tid; idx < 16*128; idx += 256) {
                int r = idx >> 7, x = idx & 127;
                gacc[r][x] += buf[1][r + 5][x + 5];
            }
            __syncthreads();
        }
    }

    float* gp = gates + (((size_t)(b*4 + g) * H) + y0) * W;
    for (int idx = tid; idx < 16*128; idx += 256) {
        int r = idx >> 7, x = idx & 127;
        gp[r*W + x] = gacc[r][x];
    }
}

// ================== Phase 2: dense 16->16 3x3 conv via f32 WMMA ==================
// grid: (128 rows, 32 batch), block 256 = 8 waves; wave w owns x-strip [16w, 16w+16).
// GEMM: M = 16 pixels, N = 16 out-channels, K = 4 in-channel chunk;
// 9 taps x 4 chunks = 36 v_wmma_f32_16x16x4_f32 accumulations per tile.
// mode 0: out = conv+bias ; mode 1: out = (conv+bias)*gate ; mode 2: out += (conv+bias)*gate
__global__ __launch_bounds__(256)
void conv16_wmma_kernel(const float* __restrict__ in,
                        float* __restrict__ out,
                        const float* __restrict__ w,
                        const float* __restrict__ bias,
                        const float* __restrict__ gates,
                        int gidx, int mode)
{
    const int y   = blockIdx.x;
    const int b   = blockIdx.y;
    const int tid = threadIdx.x;

    __shared__ float inT[3*130*18];   // [row 0..2][xh 0..129][ch, padded 16->18]
    __shared__ v2f   wB[36*32];       // [tap*4+kc][lane] in WMMA B-operand layout

    // stage input rows y-1..y+1, all 16 channels, x=-1..128 with zero padding.
    // linear order (ch, r, xh) so consecutive threads read consecutive x (coalesced).
    for (int idx = tid; idx < 16*3*130; idx += 256) {
        int ch  = idx / 390;
        int rem = idx - ch*390;
        int r   = rem / 130;
        int xh  = rem - r*130;
        int yi = y + r - 1;
        int x  = xh - 1;
        float v = 0.f;
        if (yi >= 0 && yi < H && x >= 0 && x < W)
            v = in[((size_t)(b*16 + ch) * H + yi) * W + x];
        inT[(r*130 + xh)*18 + ch] = v;
    }
    // pre-pack weights into B layout: lane<16 -> (K=0,1,N=lane); lane>=16 -> (K=2,3,N=lane-16)
    for (int idx = tid; idx < 36*32; idx += 256) {
        int it   = idx >> 5;          // tap*4 + kc
        int lane = idx & 31;
        int kc  = it & 3;
        int tap = it >> 2;
        int ky = tap / 3, kx = tap - ky*3;
        int oc  = lane & 15;
        int icb = kc*4 + ((lane >> 4) << 1);
        v2f wv;
        wv.x = w[((oc*16 + icb    )*3 + ky)*3 + kx];
        wv.y = w[((oc*16 + icb + 1)*3 + ky)*3 + kx];
        wB[idx] = wv;
    }
    __syncthreads();

    const int lane  = tid & 31;
    const int wave  = tid >> 5;
    const int x0    = wave * 16;
    const int laneM = lane & 15;
    const int half  = lane >> 4;

    v8f acc = {0.f,0.f,0.f,0.f,0.f,0.f,0.f,0.f};
    #pragma unroll
    for (int ky = 0; ky < 3; ++ky) {
        #pragma unroll
        for (int kx = 0; kx < 3; ++kx) {
            const int xh = x0 + laneM + kx;                       // x + dx, shifted by +1
            const float* ap = &inT[(ky*130 + xh)*18 + 2*half];    // 8B-aligned pair base
            const int tap = ky*3 + kx;
            #pragma unroll
            for (int kc = 0; kc < 4; ++kc) {
                v2f a;
                a.x = ap[kc*4];        // A[M=laneM, K = 2*half + 0]  (ic = 4*kc+2*half)
                a.y = ap[kc*4 + 1];    // A[M=laneM, K = 2*half + 1]
                v2f bb = wB[(tap*4 + kc)*32 + lane];
                acc = __builtin_amdgcn_wmma_f32_16x16x4_f32(
                          false, a, false, bb, (short)0, acc, false, false);
            }
        }
    }

    // D layout: lane<16 -> channel=lane, pixels x0+0..7 ; lane>=16 -> channel=lane-16, pixels x0+8..15
    const float bv = bias[laneM];
    const int xb = x0 + half*8;
    float* op = out + ((size_t)(b*16 + laneM) * H + y) * W + xb;
    if (mode == 0) {
        #pragma unroll
        for (int r = 0; r < 8; ++r) op[r] = acc[r] + bv;
    } else {
        const float* gp = gates + ((size_t)(b*4 + gidx) * H + y) * W + xb;
        #pragma unroll
        for (int r = 0; r < 8; ++r) {
            float v = (acc[r] + bv) * gp[r];
            if (mode == 2) v += op[r];
            op[r] = v;
        }
    }
}

// ==================================== launch ====================================
extern "C" void kernel_launch(void* const* d_in, const int* in_sizes, int n_in,
                              void* d_out, int out_size, void* d_ws, size_t ws_size,
                              hipStream_t stream)
{
    (void)in_sizes; (void)n_in; (void)out_size; (void)ws_size;
    const float* landmarks = (const float*)d_in[0];  // [32,68,128,128]
    const float* features  = (const float*)d_in[1];  // [32,16,128,128]
    const float* w_lm      = (const float*)d_in[2];  // [5,68,1,3,3]
    const float* b_lm      = (const float*)d_in[3];  // [5,68]
    const float* w_fea     = (const float*)d_in[4];  // [4,3,16,16,3,3]
    const float* b_fea     = (const float*)d_in[5];  // [4,3,16]
    float* out = (float*)d_out;                      // [32,16,128,128]

    char* ws = (char*)d_ws;
    float* gates = (float*)ws;                            // 32*4*128*128 f32 = 8 MiB
    float* bufA  = (float*)(ws + (size_t)8  * 1024*1024); // 32 MiB ping
    float* bufB  = (float*)(ws + (size_t)40 * 1024*1024); // 32 MiB pong

    dw_gate_kernel<<<dim3(8, 4, 32), 256, 0, stream>>>(landmarks, w_lm, b_lm, gates);

    for (int g = 0; g < 4; ++g) {
        const float* w0  = w_fea + (size_t)(g*3 + 0) * 16*16*9;
        const float* w1  = w_fea + (size_t)(g*3 + 1) * 16*16*9;
        const float* w2  = w_fea + (size_t)(g*3 + 2) * 16*16*9;
        const float* bb0 = b_fea + (g*3 + 0) * 16;
        const float* bb1 = b_fea + (g*3 + 1) * 16;
        const float* bb2 = b_fea + (g*3 + 2) * 16;
        conv16_wmma_kernel<<<dim3(128, 32), 256, 0, stream>>>(features, bufA, w0, bb0, gates, g, 0);
        conv16_wmma_kernel<<<dim3(128, 32), 256, 0, stream>>>(bufA,     bufB, w1, bb1, gates, g, 0);
        conv16_wmma_kernel<<<dim3(128, 32), 256, 0, stream>>>(bufB,     out,  w2, bb2, gates, g, (g == 0) ? 1 : 2);
    }
}